// VisionMamba_70291434766728
// MI455X (gfx1250) — compile-verified
//
#include <hip/hip_runtime.h>
#include <math.h>

// ---------------------------------------------------------------------------
// VisionMamba forward for MI455X (gfx1250), f32 end-to-end (matches reference).
//
// Dims: B=32, C=3, IMG=384, PATCH=8 -> h=w=48, N=2304 patches, D=192 embed.
// Rows = B*N = 73728.
//
// Stage 1: patch-embed GEMM (73728x192)@(192x192) via V_WMMA_F32_16X16X4_F32,
//          one wave32 per 16x16 output tile, A gathered on the fly from x.
// Stage 2: LayerNorm + dw affine + channel mean -> u[b,n]  (wave per row).
// Stage 3: dA = exp(delta*A) (stored transposed), then the sequential scan
//          w_{k+1} = dA @ w_k ; kernel[k] = C . w_k  (single workgroup).
// Stage 4: direct causal conv fwd/bwd; u row staged into LDS via
//          GLOBAL_LOAD_ASYNC_TO_LDS_B128 (ASYNCcnt), kernel taps staged via
//          the Tensor Data Mover (TENSORcnt) when the builtins are available.
// Stage 5: fused residual + LN2 + un-patchify + 1x1 merge conv -> output.
// ---------------------------------------------------------------------------

typedef __attribute__((ext_vector_type(2))) float        v2f;
typedef __attribute__((ext_vector_type(8))) float        v8f;
typedef __attribute__((ext_vector_type(4))) unsigned int v4u;
typedef __attribute__((ext_vector_type(8))) int          v8i;
typedef __attribute__((ext_vector_type(4))) int          v4i;

#define AS1 __attribute__((address_space(1)))
#define AS3 __attribute__((address_space(3)))

#if defined(__has_builtin)
#  if __has_builtin(__builtin_amdgcn_global_load_async_to_lds_b128)
#    define HAVE_ASYNC_LDS 1
#  endif
#  if __has_builtin(__builtin_amdgcn_tensor_load_to_lds) && \
      __has_builtin(__builtin_amdgcn_s_wait_tensorcnt)
#    define HAVE_TDM 1
#  endif
#endif
#ifndef HAVE_ASYNC_LDS
#  define HAVE_ASYNC_LDS 0
#endif
#ifndef HAVE_TDM
#  define HAVE_TDM 0
#endif

#define NBATCH   32
#define NPATCH   2304      // 48*48
#define NROWS    73728     // NBATCH*NPATCH
#define DEMB     192
#define IMGHW    147456    // 384*384

// ---------------- Stage 1: WMMA patch-embed GEMM ---------------------------
// grid (576, 12): blockIdx.x*8+wave = row-tile (4608), blockIdx.y = col-tile.
__global__ __launch_bounds__(256) void k_patch_gemm(
    const float* __restrict__ x, const float* __restrict__ proj_w,
    const float* __restrict__ proj_b, const float* __restrict__ pos,
    float* __restrict__ patches)
{
    const int tile_m = blockIdx.x * 8 + (threadIdx.x >> 5);
    const int tile_n = blockIdx.y;
    const int lane   = threadIdx.x & 31;
    const int half   = lane >> 4;   // 0: K pair {0,1}, 1: K pair {2,3}
    const int l16    = lane & 15;

    const int r0  = tile_m * 16;    // first output row of tile
    const int n0  = tile_n * 16;    // first output col of tile
    const int bb  = r0 / NPATCH;    // 2304 % 16 == 0 -> tile stays in one batch
    const int np0 = r0 % NPATCH;    // patch index of tile row 0

    // A-operand row for this lane (M = l16), gathered straight from x.
    const int np = np0 + l16;
    const int hh = np / 48, ww = np % 48;
    const float* xb = x + (size_t)bb * (3 * IMGHW) + (size_t)(hh * 8) * 384 + (ww * 8);

    // B-operand column for this lane (N = l16): Wk[k][d] = proj_w[d*192 + k].
    const float* wd = proj_w + (size_t)(n0 + l16) * DEMB;

    v8f acc = {};
    for (int k0 = 0; k0 < DEMB; k0 += 4) {
        const int ka = k0 + half * 2;
        const int kb = ka + 1;
        // k -> (c,p,q): c=k/64, p=(k%64)/8, q=k%8
        const int c0 = ka >> 6, p0 = (ka >> 3) & 7, q0 = ka & 7;
        const int c1 = kb >> 6, p1 = (kb >> 3) & 7, q1 = kb & 7;
        v2f a, b;
        a.x = xb[c0 * IMGHW + p0 * 384 + q0];
        a.y = xb[c1 * IMGHW + p1 * 384 + q1];
        b.x = wd[ka];
        b.y = wd[kb];
        acc = __builtin_amdgcn_wmma_f32_16x16x4_f32(
            /*neg_a=*/false, a, /*neg_b=*/false, b,
            /*c_mod=*/(short)0, acc, /*reuse_a=*/false, /*reuse_b=*/false);
    }

    // C/D layout: VGPR v -> M = v + half*8, N = l16.
    const int   col  = n0 + l16;
    const float bias = proj_b[col];
    #pragma unroll
    for (int v = 0; v < 8; ++v) {
        const int rr  = r0 + half * 8 + v;
        const int npp = np0 + half * 8 + v;
        patches[(size_t)rr * DEMB + col] = acc[v] + bias + pos[(size_t)npp * DEMB + col];
    }
}

// ---------------- Stage 2: LN + dw affine + channel mean -> u --------------
__global__ __launch_bounds__(256) void k_ln_u(
    const float* __restrict__ patches, const float* __restrict__ ln_w,
    const float* __restrict__ ln_b, const float* __restrict__ dw_w,
    const float* __restrict__ dw_b, float* __restrict__ u)
{
    const int wid  = blockIdx.x * 8 + (threadIdx.x >> 5);  // row id
    const int lane = threadIdx.x & 31;
    const float* row = patches + (size_t)wid * DEMB;

    float v0[6];
    float sum = 0.f;
    #pragma unroll
    for (int i = 0; i < 6; ++i) { v0[i] = row[lane + i * 32]; sum += v0[i]; }
    #pragma unroll
    for (int off = 16; off > 0; off >>= 1) sum += __shfl_xor(sum, off, 32);
    const float mu = sum * (1.0f / DEMB);

    float var = 0.f;
    #pragma unroll
    for (int i = 0; i < 6; ++i) { const float d = v0[i] - mu; var += d * d; }
    #pragma unroll
    for (int off = 16; off > 0; off >>= 1) var += __shfl_xor(var, off, 32);
    const float rstd = rsqrtf(var * (1.0f / DEMB) + 1e-5f);

    float acc = 0.f;
    #pragma unroll
    for (int i = 0; i < 6; ++i) {
        const int d = lane + i * 32;
        const float ln = (v0[i] - mu) * rstd * ln_w[d] + ln_b[d];
        acc += ln * dw_w[d] + dw_b[d];
    }
    #pragma unroll
    for (int off = 16; off > 0; off >>= 1) acc += __shfl_xor(acc, off, 32);
    if (lane == 0) u[wid] = acc * (1.0f / DEMB);
}

// ---------------- Stage 3a: dA = exp(delta*A), stored transposed -----------
__global__ __launch_bounds__(256) void k_expA(
    const float* __restrict__ A, const float* __restrict__ delta,
    float* __restrict__ dAT)
{
    const int t = blockIdx.x * 256 + threadIdx.x;
    if (t < DEMB * DEMB) {
        const int d = t / DEMB, j = t % DEMB;
        dAT[j * DEMB + d] = expf(delta[0] * A[t]);   // dAT[j][d] = dA[d][j]
    }
}

// ---------------- Stage 3b: sequential scan for kernel taps ----------------
__global__ __launch_bounds__(192) void k_gen_kernel(
    const float* __restrict__ dAT, const float* __restrict__ Bv,
    const float* __restrict__ Cv, const float* __restrict__ delta,
    float* __restrict__ kern)
{
    __shared__ float w[DEMB];
    __shared__ float red[DEMB];
    const int t = threadIdx.x;
    w[t] = delta[0] * Bv[t];          // w_0 = delta * B
    const float c = Cv[t];
    __syncthreads();

    for (int k = 0; k < NPATCH; ++k) {
        red[t] = c * w[t];
        __syncthreads();
        if (t < 96) red[t] += red[t + 96];
        __syncthreads();
        if (t < 48) red[t] += red[t + 48];
        __syncthreads();
        if (t < 24) red[t] += red[t + 24];
        __syncthreads();
        if (t < 12) red[t] += red[t + 12];
        __syncthreads();
        if (t < 6)  red[t] += red[t + 6];
        __syncthreads();
        if (t == 0)
            kern[k] = red[0] + red[1] + red[2] + red[3] + red[4] + red[5];

        // w_{k+1}[t] = sum_j dA[t][j]*w[j] = sum_j dAT[j][t]*w[j] (coalesced)
        float acc = 0.f;
        for (int j = 0; j < DEMB; ++j) acc += dAT[j * DEMB + t] * w[j];
        __syncthreads();
        w[t] = acc;
        __syncthreads();
    }
}

// ---------------- Stage 4: direct causal conv (fwd + bwd) ------------------
// u row staged via async global->LDS (ASYNCcnt); kernel taps via TDM
// (TENSORcnt) when available, with plain-LDS fallbacks otherwise.
__global__ __launch_bounds__(256) void k_conv(
    const float* __restrict__ u, const float* __restrict__ kern,
    float* __restrict__ s)
{
    __shared__ float su[NPATCH];
    __shared__ float sk[NPATCH];
    const int b = blockIdx.x;
    const float* gu = u + (size_t)b * NPATCH;

#if HAVE_ASYNC_LDS
    {
        AS1 v4i* gsrc = (AS1 v4i*)gu;     // global source, 16B granules
        AS3 v4i* ldst = (AS3 v4i*)su;     // LDS destination, 16B granules
        for (int i = threadIdx.x; i < NPATCH / 4; i += 256)   // 576 x b128
            __builtin_amdgcn_global_load_async_to_lds_b128(
                gsrc + i, ldst + i, 0, 0);
    }
#else
    for (int i = threadIdx.x; i < NPATCH; i += 256) su[i] = gu[i];
#endif

#if HAVE_TDM
    if (threadIdx.x < 32) {   // wave 0 issues one TDM descriptor for sk[]
        const unsigned long long ga = (unsigned long long)(size_t)kern;
        const unsigned int       la = (unsigned int)(size_t)(AS3 char*)sk;
        v4u g0;
        g0.x = 1u;                                          // count=1, user D#
        g0.y = la;                                          // lds_addr
        g0.z = (unsigned int)(ga & 0xffffffffull);          // global_addr lo
        g0.w = (unsigned int)((ga >> 32) & 0x01ffffffull)   // global_addr hi
               | (2u << 30);                                // type = 2 (image)
        v8i g1;
        g1[0] = (int)(2u << 16);                            // data_size = 4B
        g1[1] = (int)((NPATCH & 0xffffu) << 16);            // tensor_dim0 lo16
        g1[2] = (int)(((NPATCH >> 16) & 0xffffu)            // tensor_dim0 hi16
                      | (1u << 16));                        // tensor_dim1 = 1
        g1[3] = (int)((NPATCH & 0xffffu) << 16);            // tile_dim0 = 2304
        g1[4] = 1;                                          // tile_dim1 = 1
        g1[5] = NPATCH;                                     // dim0_stride lo32
        g1[6] = (int)((NPATCH & 0xffffu) << 16);            // dim1_stride lo16
        g1[7] = 0;
        v4i g2 = {0, 0, 0, 0};
        v4i g3 = {0, 0, 0, 0};
#if defined(__clang_major__) && (__clang_major__ >= 23)
        v8i gz = {0, 0, 0, 0, 0, 0, 0, 0};
        __builtin_amdgcn_tensor_load_to_lds(g0, g1, g2, g3, gz, 0);
#else
        __builtin_amdgcn_tensor_load_to_lds(g0, g1, g2, g3, 0);
#endif
        __builtin_amdgcn_s_wait_tensorcnt(0);
    }
#else
    for (int i = threadIdx.x; i < NPATCH; i += 256) sk[i] = kern[i];
#endif

#if HAVE_ASYNC_LDS
#  if defined(__has_builtin) && __has_builtin(__builtin_amdgcn_s_wait_asynccnt)
    __builtin_amdgcn_s_wait_asynccnt(0);
#  else
    asm volatile("s_wait_asynccnt 0" ::: "memory");
#  endif
#endif
    __syncthreads();

    for (int n = threadIdx.x; n < NPATCH; n += 256) {
        float f = 0.f, bw = 0.f;
        for (int k = 0; k <= n; ++k) {
            const float uk = su[k];
            f  += uk * sk[n - k];               // fwd:  kernel[n-k]
            bw += uk * sk[NPATCH - 1 - n + k];  // bwd:  kernel[N-1-(n-k)]
        }
        s[(size_t)b * NPATCH + n] = 0.5f * (f + bw);
    }
}

// ---------------- Stage 5: residual + LN2 + un-patchify + merge ------------
__global__ __launch_bounds__(256) void k_final(
    const float* __restrict__ patches, const float* __restrict__ s,
    const float* __restrict__ ln_w, const float* __restrict__ ln_b,
    const float* __restrict__ mw, const float* __restrict__ mb,
    float* __restrict__ out)
{
    __shared__ float ybuf[8][DEMB];
    const int wslot = threadIdx.x >> 5;
    const int wid   = blockIdx.x * 8 + wslot;   // row id
    const int lane  = threadIdx.x & 31;
    const float* row = patches + (size_t)wid * DEMB;
    const float sv = s[wid];

    float v0[6];
    float sum = 0.f;
    #pragma unroll
    for (int i = 0; i < 6; ++i) { v0[i] = row[lane + i * 32]; sum += v0[i]; }
    #pragma unroll
    for (int off = 16; off > 0; off >>= 1) sum += __shfl_xor(sum, off, 32);
    const float mu = sum * (1.0f / DEMB);
    float var = 0.f;
    #pragma unroll
    for (int i = 0; i < 6; ++i) { const float d = v0[i] - mu; var += d * d; }
    #pragma unroll
    for (int off = 16; off > 0; off >>= 1) var += __shfl_xor(var, off, 32);
    const float rstd = rsqrtf(var * (1.0f / DEMB) + 1e-5f);

    // y1 = patches + s (per-row constant) => centered values/variance reused.
    #pragma unroll
    for (int i = 0; i < 6; ++i) {
        const int d = lane + i * 32;
        const float y1 = v0[i] + sv;
        const float ln = (v0[i] - mu) * rstd * ln_w[d] + ln_b[d];
        ybuf[wslot][d] = ln + y1;
    }
    __syncthreads();

    const int b  = wid / NPATCH;
    const int np = wid % NPATCH;
    const int hh = np / 48, ww = np % 48;
    #pragma unroll
    for (int i = 0; i < 6; ++i) {
        const int e    = lane + i * 32;          // 192 outputs per row
        const int dcol = e >> 6;                 // output channel 0..2
        const int pp   = (e >> 3) & 7;
        const int qq   = e & 7;
        float acc = mb[dcol];
        #pragma unroll
        for (int cc = 0; cc < 3; ++cc)
            acc += ybuf[wslot][pp * 24 + qq * 3 + cc] * mw[dcol * 3 + cc];
        out[((size_t)b * 3 + dcol) * IMGHW + (size_t)(hh * 8 + pp) * 384 + (ww * 8 + qq)] = acc;
    }
}

// ---------------------------------------------------------------------------
extern "C" void kernel_launch(void* const* d_in, const int* in_sizes, int n_in,
                              void* d_out, int out_size, void* d_ws, size_t ws_size,
                              hipStream_t stream) {
    const float* x      = (const float*)d_in[0];
    const float* proj_w = (const float*)d_in[1];
    const float* proj_b = (const float*)d_in[2];
    const float* pos    = (const float*)d_in[3];
    const float* ln_w   = (const float*)d_in[4];
    const float* ln_b   = (const float*)d_in[5];
    const float* ssm_A  = (const float*)d_in[6];
    const float* ssm_B  = (const float*)d_in[7];
    const float* ssm_C  = (const float*)d_in[8];
    const float* ssm_dl = (const float*)d_in[9];
    const float* dw_w   = (const float*)d_in[10];
    const float* dw_b   = (const float*)d_in[11];
    const float* mw     = (const float*)d_in[12];
    const float* mb     = (const float*)d_in[13];
    float* out = (float*)d_out;

    float* ws      = (float*)d_ws;
    float* patches = ws;                                   // 73728*192
    float* u       = patches + (size_t)NROWS * DEMB;       // 73728
    float* dAT     = u + NROWS;                            // 192*192
    float* kern    = dAT + DEMB * DEMB;                    // 2304
    float* s       = kern + NPATCH;                        // 73728

    k_patch_gemm<<<dim3(576, 12), 256, 0, stream>>>(x, proj_w, proj_b, pos, patches);
    k_expA      <<<144,  256, 0, stream>>>(ssm_A, ssm_dl, dAT);
    k_ln_u      <<<9216, 256, 0, stream>>>(patches, ln_w, ln_b, dw_w, dw_b, u);
    k_gen_kernel<<<1,    192, 0, stream>>>(dAT, ssm_B, ssm_C, ssm_dl, kern);
    k_conv      <<<NBATCH, 256, 0, stream>>>(u, kern, s);
    k_final     <<<9216, 256, 0, stream>>>(patches, s, ln_w, ln_b, mw, mb, out);
}